// MyModel_86835648790651
// MI455X (gfx1250) — compile-verified
//
#include <hip/hip_runtime.h>
#include <math.h>

#define T_DIM 128
#define N_DIM 1024
#define F_INP 7
#define F_OUT 7
#define U1 5
#define CF 5
#define CK 5
#define DEC_IN 15   // 2*U1 + CF
#define CCS 16      // padded stride for concat features

typedef float v2f __attribute__((ext_vector_type(2)));
typedef float v8f __attribute__((ext_vector_type(8)));

__device__ __forceinline__ float sigm(float x) { return 1.0f / (1.0f + __expf(-x)); }

// ---------------------------------------------------------------------------
// K1: h = X@W ; s1 = h@a1 ; s2 = h@a2   (per (t,n))
// ---------------------------------------------------------------------------
__global__ void k_proj(const float* __restrict__ X, const float* __restrict__ W,
                       const float* __restrict__ a1, const float* __restrict__ a2,
                       float* __restrict__ hlin, float* __restrict__ s1,
                       float* __restrict__ s2) {
  int idx = blockIdx.x * blockDim.x + threadIdx.x;   // t*N + n
  if (idx >= T_DIM * N_DIM) return;
  const float* x = X + (size_t)idx * F_INP;
  float xv[F_INP];
#pragma unroll
  for (int f = 0; f < F_INP; ++f) xv[f] = x[f];
  float acc1 = 0.f, acc2 = 0.f;
#pragma unroll
  for (int o = 0; o < F_OUT; ++o) {
    float h = 0.f;
#pragma unroll
    for (int f = 0; f < F_INP; ++f) h += xv[f] * W[f * F_OUT + o];
    hlin[(size_t)idx * F_OUT + o] = h;
    acc1 += h * a1[o];
    acc2 += h * a2[o];
  }
  s1[idx] = acc1;
  s2[idx] = acc2;
}

// ---------------------------------------------------------------------------
// K2: GAT attention + aggregation via V_WMMA_F32_16X16X4_F32.
// One wave per 16-row tile, 8 tiles per block (128 rows), one t per block.x.
// p_ij = A_ij * exp(lrelu(s1_i + s2_j)) feeds the A-fragment from registers;
// the B-fragment comes from LDS: s_h[j][0..6] = h_j, s_h[j][7] = 1.0 so the
// accumulator's column 7 collects the softmax denominator. C columns 8..15
// are dead, so lanes n>=8 simply mirror lanes n-8 (no select, no EXEC churn).
// ---------------------------------------------------------------------------
__global__ void __launch_bounds__(256) k_gat_attn(
    const float* __restrict__ A, const float* __restrict__ hlin,
    const float* __restrict__ s1, const float* __restrict__ s2,
    const float* __restrict__ gat_b, float* __restrict__ H) {
  __shared__ float s_s2[N_DIM];        // 4 KB
  __shared__ float s_h[N_DIM * 8];     // 32 KB: cols 0..6 = h, col 7 = 1.0

  const int t = blockIdx.x;
  const int tid = threadIdx.x;
  const float* __restrict__ s2t = s2 + (size_t)t * N_DIM;
  const float* __restrict__ hb = hlin + (size_t)t * N_DIM * F_OUT;

  // Stage s2 and padded h for this timestep (shared by all 8 waves).
#pragma unroll
  for (int i = 0; i < N_DIM / 256; ++i) s_s2[tid + i * 256] = s2t[tid + i * 256];
#pragma unroll
  for (int i = 0; i < (N_DIM * 8) / 256; ++i) {
    const int idx = tid + i * 256;
    const int j = idx >> 3;
    const int f = idx & 7;
    s_h[idx] = (f < F_OUT) ? hb[j * F_OUT + f] : 1.0f;
  }
  __syncthreads();

  const int wave = tid >> 5;
  const int lane = tid & 31;
  const int i0 = blockIdx.y * 128 + wave * 16;   // tile's first row
  const int m = lane & 15;                       // A-frag row / C col
  const int half = lane >> 4;
  const int nb = m & 7;                          // B-frag LDS column

  const float si = s1[t * N_DIM + i0 + m];
  const float* __restrict__ Arow = A + (size_t)(i0 + m) * N_DIM;

  v8f c = {0.f, 0.f, 0.f, 0.f, 0.f, 0.f, 0.f, 0.f};

  for (int jb = 0; jb < N_DIM; jb += 128) {
    if (jb + 128 < N_DIM) __builtin_prefetch(Arow + jb + 128, 0, 1);
#pragma unroll 8
    for (int jc = jb; jc < jb + 128; jc += 4) {
      const int j0 = jc + 2 * half;  // lane's K pair: j0, j0+1

      float e0 = si + s_s2[j0];
      float e1 = si + s_s2[j0 + 1];
      e0 = fmaxf(e0, 0.f) + 0.2f * fminf(e0, 0.f);   // leaky_relu(0.2)
      e1 = fmaxf(e1, 0.f) + 0.2f * fminf(e1, 0.f);

      v2f a;
      a.x = Arow[j0] * __expf(e0);       // masked unnormalized attention
      a.y = Arow[j0 + 1] * __expf(e1);

      v2f b;
      b.x = s_h[(j0 << 3) + nb];
      b.y = s_h[((j0 + 1) << 3) + nb];

      c = __builtin_amdgcn_wmma_f32_16x16x4_f32(false, a, false, b,
                                                (short)0, c, false, false);
    }
  }

  // C/D layout: register r -> row (r + 8*half), column (lane & 15).
  // Denominator (col 7) lives in lane half*16+7 at the same register.
#pragma unroll
  for (int r = 0; r < 8; ++r) {
    const int mr = r + 8 * half;
    const float den = __shfl(c[r], half * 16 + 7, 32);
    if (m < F_OUT) {
      float val = c[r] / den + gat_b[m];
      val = (val > 0.f) ? val : expm1f(val);       // elu
      H[((size_t)t * N_DIM + i0 + mr) * F_OUT + m] = val;
    }
  }
}

// ---------------------------------------------------------------------------
// K3: encoder bidirectional GRU (hidden U1=5), one thread per (node, dir).
// dir is uniform per block (blocks 0..3 fwd, 4..7 bwd). Weights in LDS.
// Writes fwd into cc[...,0:5], bwd into cc[...,5:10].
// ---------------------------------------------------------------------------
__global__ void __launch_bounds__(256) k_enc_gru(
    const float* __restrict__ H,
    const float* __restrict__ gf_Wx, const float* __restrict__ gf_Wh,
    const float* __restrict__ gf_b, const float* __restrict__ gb_Wx,
    const float* __restrict__ gb_Wh, const float* __restrict__ gb_b,
    float* __restrict__ cc) {
  __shared__ float sWx[F_OUT * 3 * U1];   // 105
  __shared__ float sWh[U1 * 3 * U1];      // 75
  __shared__ float sb[2 * 3 * U1];        // 30

  const int gid = blockIdx.x * blockDim.x + threadIdx.x;  // 0..2047
  const int dir = (gid >= N_DIM) ? 1 : 0;
  const int nn = gid & (N_DIM - 1);
  const float* Wx = dir ? gb_Wx : gf_Wx;
  const float* Wh = dir ? gb_Wh : gf_Wh;
  const float* bb = dir ? gb_b : gf_b;

  if (threadIdx.x < F_OUT * 3 * U1) sWx[threadIdx.x] = Wx[threadIdx.x];
  if (threadIdx.x < U1 * 3 * U1) sWh[threadIdx.x] = Wh[threadIdx.x];
  if (threadIdx.x < 2 * 3 * U1) sb[threadIdx.x] = bb[threadIdx.x];
  __syncthreads();

  float h[U1] = {0.f, 0.f, 0.f, 0.f, 0.f};
  for (int s = 0; s < T_DIM; ++s) {
    const int t = dir ? (T_DIM - 1 - s) : s;
    const float* x = H + ((size_t)t * N_DIM + nn) * F_OUT;
    float gx[3 * U1], gh[3 * U1];
#pragma unroll
    for (int g = 0; g < 3 * U1; ++g) { gx[g] = sb[g]; gh[g] = sb[3 * U1 + g]; }
#pragma unroll
    for (int f = 0; f < F_OUT; ++f) {
      const float xv = x[f];
#pragma unroll
      for (int g = 0; g < 3 * U1; ++g) gx[g] += xv * sWx[f * 3 * U1 + g];
    }
#pragma unroll
    for (int u = 0; u < U1; ++u) {
      const float hv = h[u];
#pragma unroll
      for (int g = 0; g < 3 * U1; ++g) gh[g] += hv * sWh[u * 3 * U1 + g];
    }
    float* o = cc + ((size_t)t * N_DIM + nn) * CCS + dir * U1;
#pragma unroll
    for (int u = 0; u < U1; ++u) {
      const float z = sigm(gx[u] + gh[u]);
      const float r = sigm(gx[U1 + u] + gh[U1 + u]);
      const float hc = tanhf(gx[2 * U1 + u] + r * gh[2 * U1 + u]);
      h[u] = z * h[u] + (1.f - z) * hc;
      o[u] = h[u];
    }
  }
}

// ---------------------------------------------------------------------------
// K4: depth-5 "SAME" conv along time per node; writes cc[...,10:15].
// ---------------------------------------------------------------------------
__global__ void k_conv(const float* __restrict__ H, const float* __restrict__ ck,
                       const float* __restrict__ cb, float* __restrict__ cc) {
  const int idx = blockIdx.x * blockDim.x + threadIdx.x;  // t*N + n
  if (idx >= T_DIM * N_DIM) return;
  const int t = idx >> 10;
  const int nn = idx & (N_DIM - 1);
  float acc[CF];
#pragma unroll
  for (int o = 0; o < CF; ++o) acc[o] = cb[o];
#pragma unroll
  for (int k = 0; k < CK; ++k) {
    const int tt = t + k - (CK / 2);
    if (tt < 0 || tt >= T_DIM) continue;
    const float* x = H + ((size_t)tt * N_DIM + nn) * F_OUT;
#pragma unroll
    for (int fi = 0; fi < F_OUT; ++fi) {
      const float xv = x[fi];
#pragma unroll
      for (int o = 0; o < CF; ++o) acc[o] += xv * ck[(k * F_OUT + fi) * CF + o];
    }
  }
  float* out = cc + (size_t)idx * CCS + 2 * U1;
#pragma unroll
  for (int o = 0; o < CF; ++o) out[o] = acc[o];
}

// ---------------------------------------------------------------------------
// K5: decoder GRU (hidden 1). accumulate=0 writes 0.5*h, =1 adds 0.5*h.
// ---------------------------------------------------------------------------
__global__ void k_dec(const float* __restrict__ cc, const float* __restrict__ Wx,
                      const float* __restrict__ Wh, const float* __restrict__ b,
                      float* __restrict__ out, int reverse, int accumulate) {
  const int nn = blockIdx.x * blockDim.x + threadIdx.x;
  if (nn >= N_DIM) return;
  float wx[DEC_IN][3];
#pragma unroll
  for (int i = 0; i < DEC_IN; ++i)
#pragma unroll
    for (int g = 0; g < 3; ++g) wx[i][g] = Wx[i * 3 + g];
  const float wh0 = Wh[0], wh1 = Wh[1], wh2 = Wh[2];
  const float b0 = b[0], b1 = b[1], b2 = b[2];
  const float b3 = b[3], b4 = b[4], b5 = b[5];

  float h = 0.f;
  for (int s = 0; s < T_DIM; ++s) {
    const int t = reverse ? (T_DIM - 1 - s) : s;
    const float* x = cc + ((size_t)t * N_DIM + nn) * CCS;
    float gx0 = b0, gx1 = b1, gx2 = b2;
#pragma unroll
    for (int i = 0; i < DEC_IN; ++i) {
      const float xv = x[i];
      gx0 += xv * wx[i][0];
      gx1 += xv * wx[i][1];
      gx2 += xv * wx[i][2];
    }
    const float gh0 = b3 + h * wh0;
    const float gh1 = b4 + h * wh1;
    const float gh2 = b5 + h * wh2;
    const float z = sigm(gx0 + gh0);
    const float r = sigm(gx1 + gh1);
    const float hc = tanhf(gx2 + r * gh2);
    h = z * h + (1.f - z) * hc;
    const float v = 0.5f * h;
    if (accumulate)
      out[t * N_DIM + nn] += v;
    else
      out[t * N_DIM + nn] = v;
  }
}

// ---------------------------------------------------------------------------
extern "C" void kernel_launch(void* const* d_in, const int* in_sizes, int n_in,
                              void* d_out, int out_size, void* d_ws,
                              size_t ws_size, hipStream_t stream) {
  const float* X = (const float*)d_in[0];
  const float* A = (const float*)d_in[1];
  const float* gat_W = (const float*)d_in[2];
  const float* gat_a1 = (const float*)d_in[3];
  const float* gat_a2 = (const float*)d_in[4];
  const float* gat_b = (const float*)d_in[5];
  const float* gf_Wx = (const float*)d_in[6];
  const float* gf_Wh = (const float*)d_in[7];
  const float* gf_b = (const float*)d_in[8];
  const float* gb_Wx = (const float*)d_in[9];
  const float* gb_Wh = (const float*)d_in[10];
  const float* gb_b = (const float*)d_in[11];
  const float* conv_k = (const float*)d_in[12];
  const float* conv_b = (const float*)d_in[13];
  const float* df_Wx = (const float*)d_in[14];
  const float* df_Wh = (const float*)d_in[15];
  const float* df_b = (const float*)d_in[16];
  const float* db_Wx = (const float*)d_in[17];
  const float* db_Wh = (const float*)d_in[18];
  const float* db_b = (const float*)d_in[19];
  float* out = (float*)d_out;

  float* ws = (float*)d_ws;
  float* hlin = ws;                                  // T*N*7
  float* s1 = hlin + (size_t)T_DIM * N_DIM * F_OUT;  // T*N
  float* s2 = s1 + (size_t)T_DIM * N_DIM;            // T*N
  float* H = s2 + (size_t)T_DIM * N_DIM;             // T*N*7
  float* cc = H + (size_t)T_DIM * N_DIM * F_OUT;     // T*N*16

  const int TN = T_DIM * N_DIM;

  k_proj<<<(TN + 255) / 256, 256, 0, stream>>>(X, gat_W, gat_a1, gat_a2, hlin,
                                               s1, s2);
  k_gat_attn<<<dim3(T_DIM, N_DIM / 128), 256, 0, stream>>>(A, hlin, s1, s2,
                                                           gat_b, H);
  k_enc_gru<<<8, 256, 0, stream>>>(H, gf_Wx, gf_Wh, gf_b, gb_Wx, gb_Wh, gb_b,
                                   cc);
  k_conv<<<(TN + 255) / 256, 256, 0, stream>>>(H, conv_k, conv_b, cc);
  k_dec<<<N_DIM / 256, 256, 0, stream>>>(cc, df_Wx, df_Wh, df_b, out, 0, 0);
  k_dec<<<N_DIM / 256, 256, 0, stream>>>(cc, db_Wx, db_Wh, db_b, out, 1, 1);
}